// ControlFlowScanCDistXY_46308337386063
// MI455X (gfx1250) — compile-verified
//
#include <hip/hip_runtime.h>
#include <stdint.h>
#include <math.h>

typedef __attribute__((ext_vector_type(2))) float v2f;
typedef __attribute__((ext_vector_type(8))) float v8f;

#define K_DIM   1024
#define N_DIM   8192
#define BM      128
#define BN      128
#define BK      16
#define PITCH   20                       // floats per LDS row: 16 data + 4 pad (16B-aligned rows, conflict-free b64 reads)
#define NTILES  (K_DIM / BK)             // 64
#define LDS_BUF_FLOATS (BM * PITCH)      // 2560 floats = 10 KB per buffer
#define LDS_BUF_BYTES  (LDS_BUF_FLOATS * 4)

#define WMMA_F32(A, B, C) \
    __builtin_amdgcn_wmma_f32_16x16x4_f32(false, (A), false, (B), (short)0, (C), false, false)

__global__ __launch_bounds__(256)
void cdist_xy_wmma(const float* __restrict__ x,
                   const float* __restrict__ y,
                   float* __restrict__ out)
{
    // [A buf0][A buf1][B buf0][B buf1] = 40 KB
    __shared__ float lds[4 * LDS_BUF_FLOATS];

    const int tid      = threadIdx.x;
    const int lane     = tid & 31;
    const int wave     = tid >> 5;
    const int laneM    = lane & 15;      // row within 16x16 frag (A) / col (B,C)
    const int laneHalf = lane >> 4;      // 0: K 0-1 / rows 0-7 ; 1: K 2-3 / rows 8-15
    const int laneK    = laneHalf * 2;

    const int waveM = (wave & 3) * 32;   // 4 waves along M, 32 rows each
    const int waveN = (wave >> 2) * 64;  // 2 waves along N, 64 cols each

    const int M0 = blockIdx.y * BM;
    const int N0 = blockIdx.x * BN;

    // ---- async global->LDS tile fill: 2x b128 per matrix per thread per tile ----
    const int lrow = tid >> 2;           // 0..63 (two row groups: lrow, lrow+64)
    const int lc4  = tid & 3;            // float4 column within BK=16
    const unsigned gxOff0 = (unsigned)(M0 + lrow)      * (K_DIM * 4) + lc4 * 16;
    const unsigned gxOff1 = (unsigned)(M0 + lrow + 64) * (K_DIM * 4) + lc4 * 16;
    const unsigned gyOff0 = (unsigned)(N0 + lrow)      * (K_DIM * 4) + lc4 * 16;
    const unsigned gyOff1 = (unsigned)(N0 + lrow + 64) * (K_DIM * 4) + lc4 * 16;

    // low 32 bits of the flat address of an LDS object == its LDS byte address
    const unsigned ldsBase = (unsigned)(uintptr_t)(&lds[0]);
    const unsigned ldsRow0 = (unsigned)((lrow      * PITCH + lc4 * 4) * 4);
    const unsigned ldsRow1 = (unsigned)(((lrow+64) * PITCH + lc4 * 4) * 4);

    auto issue_tile = [&](int kt, int buf) {
        const unsigned kByte = (unsigned)kt * (BK * 4);
        const unsigned aDst  = ldsBase + (unsigned)buf * LDS_BUF_BYTES;
        const unsigned bDst  = ldsBase + 2u * LDS_BUF_BYTES + (unsigned)buf * LDS_BUF_BYTES;
        asm volatile("global_load_async_to_lds_b128 %0, %1, %2 offset:0"
                     :: "v"(aDst + ldsRow0), "v"(gxOff0 + kByte), "s"(x) : "memory");
        asm volatile("global_load_async_to_lds_b128 %0, %1, %2 offset:0"
                     :: "v"(aDst + ldsRow1), "v"(gxOff1 + kByte), "s"(x) : "memory");
        asm volatile("global_load_async_to_lds_b128 %0, %1, %2 offset:0"
                     :: "v"(bDst + ldsRow0), "v"(gyOff0 + kByte), "s"(y) : "memory");
        asm volatile("global_load_async_to_lds_b128 %0, %1, %2 offset:0"
                     :: "v"(bDst + ldsRow1), "v"(gyOff1 + kByte), "s"(y) : "memory");
    };

    // ---- accumulators ----
    v8f acc[2][4] = {};
    float xn[2] = {0.f, 0.f};            // partial ||x_row||^2 (this lane's K half)
    float yn[4] = {0.f, 0.f, 0.f, 0.f};  // partial ||y_col||^2

    const int aOff = (waveM + laneM) * PITCH + laneK;   // + i*16*PITCH + 4*ks
    const int bOff = (waveN + laneM) * PITCH + laneK;   // + j*16*PITCH + 4*ks

    issue_tile(0, 0);
    asm volatile("s_wait_asynccnt 0" ::: "memory");
    __syncthreads();

    for (int kt = 0; kt < NTILES; ++kt) {
        const int cur = kt & 1;
        if (kt + 1 < NTILES) issue_tile(kt + 1, cur ^ 1);

        // runtime offset select (NO constant pointer arrays -> no addrspacecast initializers)
        const int bufOff = cur * LDS_BUF_FLOATS;
        const float* pa = lds + bufOff;
        const float* pb = lds + 2 * LDS_BUF_FLOATS + bufOff;

        #pragma unroll
        for (int ks = 0; ks < BK / 4; ++ks) {
            const int kc = ks * 4;
            v2f a0 = *(const v2f*)(pa + aOff + kc);
            v2f a1 = *(const v2f*)(pa + aOff + 16 * PITCH + kc);
            v2f b0 = *(const v2f*)(pb + bOff + kc);
            v2f b1 = *(const v2f*)(pb + bOff + 16 * PITCH + kc);
            v2f b2 = *(const v2f*)(pb + bOff + 32 * PITCH + kc);
            v2f b3 = *(const v2f*)(pb + bOff + 48 * PITCH + kc);

            acc[0][0] = WMMA_F32(a0, b0, acc[0][0]);
            acc[0][1] = WMMA_F32(a0, b1, acc[0][1]);
            acc[0][2] = WMMA_F32(a0, b2, acc[0][2]);
            acc[0][3] = WMMA_F32(a0, b3, acc[0][3]);
            acc[1][0] = WMMA_F32(a1, b0, acc[1][0]);
            acc[1][1] = WMMA_F32(a1, b1, acc[1][1]);
            acc[1][2] = WMMA_F32(a1, b2, acc[1][2]);
            acc[1][3] = WMMA_F32(a1, b3, acc[1][3]);

            xn[0] += a0.x * a0.x + a0.y * a0.y;
            xn[1] += a1.x * a1.x + a1.y * a1.y;
            yn[0] += b0.x * b0.x + b0.y * b0.y;
            yn[1] += b1.x * b1.x + b1.y * b1.y;
            yn[2] += b2.x * b2.x + b2.y * b2.y;
            yn[3] += b3.x * b3.x + b3.y * b3.y;
        }

        asm volatile("s_wait_asynccnt 0" ::: "memory");
        __syncthreads();
    }

    // ---- combine K-halves of the norms: lane L and L^16 hold the two halves of the same row/col ----
    #pragma unroll
    for (int i = 0; i < 2; ++i) xn[i] += __shfl_xor(xn[i], 16, 32);
    #pragma unroll
    for (int j = 0; j < 4; ++j) yn[j] += __shfl_xor(yn[j], 16, 32);
    // now lane L holds full ||x_{fragRow (L&15)}||^2 and ||y_{fragCol (L&15)}||^2

    // ---- epilogue: out = sqrt(max(x2 + y2 - 2*xy, 0)) ----
    #pragma unroll
    for (int i = 0; i < 2; ++i) {
        #pragma unroll
        for (int j = 0; j < 4; ++j) {
            const int gm = M0 + waveM + i * 16;
            const int gn = N0 + waveN + j * 16 + laneM;
            float* po = out + (size_t)gm * N_DIM + gn;
            const float y2 = yn[j];                       // this lane's column norm
            #pragma unroll
            for (int v = 0; v < 8; ++v) {
                const int m = v + 8 * laneHalf;           // C-layout row for VGPR v
                const float x2 = __shfl(xn[i], m, 32);    // bpermute row norm into place
                const float d2 = x2 + y2 - 2.0f * acc[i][j][v];
                po[(size_t)m * N_DIM] = sqrtf(fmaxf(d2, 0.0f));
            }
        }
    }
}

extern "C" void kernel_launch(void* const* d_in, const int* in_sizes, int n_in,
                              void* d_out, int out_size, void* d_ws, size_t ws_size,
                              hipStream_t stream) {
    (void)in_sizes; (void)n_in; (void)d_ws; (void)ws_size; (void)out_size;
    const float* x = (const float*)d_in[0];
    const float* y = (const float*)d_in[1];
    float* out = (float*)d_out;
    dim3 grid(N_DIM / BN, N_DIM / BM);   // (64, 64)
    cdist_xy_wmma<<<grid, dim3(256), 0, stream>>>(x, y, out);
}